// RandomVectorQuantizer_4999341933016
// MI455X (gfx1250) — compile-verified
//
#include <hip/hip_runtime.h>

typedef float v2f __attribute__((ext_vector_type(2)));
typedef float v4f __attribute__((ext_vector_type(4)));
typedef float v8f __attribute__((ext_vector_type(8)));

#define D_DIM   64
#define TILE_C  64      // codebook columns staged per block iteration
#define CSTRIDE 34      // pairs per column row (even => halves land on odd/even bank-pairs)
#define WAVES   4
#define THREADS (WAVES * 32)

// One wave computes a 16-row M tile vs all K columns.
// dot_re = x_re.cb_re + x_im.cb_im ; dot_im = P - Q with P = x_re.cb_im, Q = x_im.cb_re
__global__ __launch_bounds__(THREADS)
void cvq_argmax_kernel(const float* __restrict__ x_re, const float* __restrict__ x_im,
                       const float* __restrict__ cb_re, const float* __restrict__ cb_im,
                       float* __restrict__ out_idx_f, int* __restrict__ ws_idx, int Kk)
{
    // lre2[c*CSTRIDE + p] = { cb_re[kc+c][2p], cb_re[kc+c][2p+1] }  (64-bit elements)
    __shared__ __attribute__((aligned(16))) v2f lre2[TILE_C * CSTRIDE];
    __shared__ __attribute__((aligned(16))) v2f lim2[TILE_C * CSTRIDE];

    const int tid  = threadIdx.x;
    const int lane = tid & 31;
    const int wave = tid >> 5;
    const int n    = lane & 15;          // column-within-subtile / M row selector
    const int half = lane >> 4;          // 0: K%4 in {0,1}; 1: K%4 in {2,3}
    const int r0   = blockIdx.x * (WAVES * 16) + wave * 16;
    const int row  = r0 + n;

    // ---- Preload A in WMMA f32 16x16x4 layout: a[t] = {A[m][4t+2h], A[m][4t+2h+1]} ----
    // t in [0,16): A = x_re ; t in [16,32): A = x_im
    v2f a[32];
    const float2* xr = (const float2*)(x_re + (size_t)row * D_DIM);
    const float2* xi = (const float2*)(x_im + (size_t)row * D_DIM);
#pragma unroll
    for (int t = 0; t < 16; ++t) {
        float2 v = xr[2 * t + half];
        v2f av; av.x = v.x; av.y = v.y;
        a[t] = av;
    }
#pragma unroll
    for (int t = 16; t < 32; ++t) {
        float2 v = xi[2 * (t - 16) + half];
        v2f av; av.x = v.x; av.y = v.y;
        a[t] = av;
    }

    float best[8];
    int   bestk[8];
#pragma unroll
    for (int i = 0; i < 8; ++i) { best[i] = -1.0f; bestk[i] = 0; }

    for (int kc = 0; kc < Kk; kc += TILE_C) {
        // ---- Stage 64x64 codebook tile pair (transposed, 128-bit packed stores) ----
#pragma unroll
        for (int j = 0; j < 8; ++j) {
            int s  = tid + j * THREADS;      // 0..1023 float4 slots
            int c  = s & 63;
            int d4 = s >> 6;                 // 0..15 -> pair rows 2*d4, 2*d4+1 at column c
            int p0 = 2 * d4;
            float4 vr = *(const float4*)(cb_re + (size_t)(kc + c) * D_DIM + 4 * d4);
            float4 vi = *(const float4*)(cb_im + (size_t)(kc + c) * D_DIM + 4 * d4);
            v4f wr; wr.x = vr.x; wr.y = vr.y; wr.z = vr.z; wr.w = vr.w;
            v4f wi; wi.x = vi.x; wi.y = vi.y; wi.z = vi.z; wi.w = vi.w;
            *(v4f*)&lre2[c * CSTRIDE + p0] = wr;   // 16B aligned: (34c + 2*d4) even
            *(v4f*)&lim2[c * CSTRIDE + p0] = wi;
        }
        // L2 prefetch of the next tile (global_prefetch_b8); uniform condition.
        if (kc + TILE_C < Kk) {
            __builtin_prefetch(cb_re + (size_t)(kc + TILE_C + (tid >> 1)) * D_DIM, 0, 1);
            __builtin_prefetch(cb_im + (size_t)(kc + TILE_C + (tid >> 1)) * D_DIM, 0, 1);
        }
        __syncthreads();

#pragma unroll 1    // keep only one subtile's accumulators live (VGPR pressure < 256)
        for (int sub = 0; sub < 4; ++sub) {
            // one shared base per tile; per-t loads are tiny immediate offsets (16*t bytes)
            const int q0 = (sub * 16 + n) * CSTRIDE + half;
            v8f accR0 = {};   // x_re . cb_re
            v8f accR1 = {};   // x_im . cb_im
            v8f accP  = {};   // x_re . cb_im
            v8f accQ  = {};   // x_im . cb_re
            // 4 independent WMMA chains (dep distance 4); t and t+16 share the LDS pairs.
#pragma unroll
            for (int t = 0; t < 16; ++t) {
                v2f br = lre2[q0 + 2 * t];
                v2f bi = lim2[q0 + 2 * t];
                accR0 = __builtin_amdgcn_wmma_f32_16x16x4_f32(
                            false, a[t],      false, br, (short)0, accR0, false, false);
                accP  = __builtin_amdgcn_wmma_f32_16x16x4_f32(
                            false, a[t],      false, bi, (short)0, accP,  false, false);
                accR1 = __builtin_amdgcn_wmma_f32_16x16x4_f32(
                            false, a[t + 16], false, bi, (short)0, accR1, false, false);
                accQ  = __builtin_amdgcn_wmma_f32_16x16x4_f32(
                            false, a[t + 16], false, br, (short)0, accQ,  false, false);
            }
            // ---- |dot|^2 and running argmax (strict > keeps earliest k) ----
            const int kcol = kc + sub * 16 + n;
#pragma unroll
            for (int i = 0; i < 8; ++i) {
                float re = accR0[i] + accR1[i];
                float im = accP[i]  - accQ[i];
                float m2 = re * re + im * im;
                if (m2 > best[i]) { best[i] = m2; bestk[i] = kcol; }
            }
        }
        __syncthreads();
    }

    // ---- Reduce across the 16 lanes holding each row; tie-break toward smaller k ----
#pragma unroll
    for (int i = 0; i < 8; ++i) {
#pragma unroll
        for (int off = 8; off >= 1; off >>= 1) {
            float om = __shfl_xor(best[i], off, 32);
            int   ok = __shfl_xor(bestk[i], off, 32);
            if (om > best[i] || (om == best[i] && ok < bestk[i])) {
                best[i] = om; bestk[i] = ok;
            }
        }
    }
    if (n == 0) {
#pragma unroll
        for (int i = 0; i < 8; ++i) {
            int rrow = r0 + half * 8 + i;    // slot i holds row M = i + 8*half
            out_idx_f[rrow] = (float)bestk[i];
            ws_idx[rrow]    = bestk[i];
        }
    }
}

// recon[n][d][0/1] = cb_re/cb_im[idx[n]][d]
__global__ __launch_bounds__(256)
void cvq_recon_kernel(const int* __restrict__ ws_idx,
                      const float* __restrict__ cb_re, const float* __restrict__ cb_im,
                      float* __restrict__ out)
{
    int g = blockIdx.x * 256 + threadIdx.x;   // [0, N*64)
    int nn = g >> 6;
    int d  = g & 63;
    int k  = ws_idx[nn];
    float re = cb_re[(size_t)k * D_DIM + d];
    float im = cb_im[(size_t)k * D_DIM + d];
    out[(size_t)nn * (2 * D_DIM) + 2 * d + 0] = re;
    out[(size_t)nn * (2 * D_DIM) + 2 * d + 1] = im;
}

extern "C" void kernel_launch(void* const* d_in, const int* in_sizes, int n_in,
                              void* d_out, int out_size, void* d_ws, size_t ws_size,
                              hipStream_t stream) {
    const float* x_re  = (const float*)d_in[0];
    const float* x_im  = (const float*)d_in[1];
    const float* cb_re = (const float*)d_in[2];
    const float* cb_im = (const float*)d_in[3];
    float* out = (float*)d_out;               // [0,N): indices as float; [N, N+N*128): recon
    int* ws_idx = (int*)d_ws;

    const int N = in_sizes[0] / D_DIM;
    const int K = in_sizes[2] / D_DIM;

    cvq_argmax_kernel<<<N / (WAVES * 16), THREADS, 0, stream>>>(
        x_re, x_im, cb_re, cb_im, out, ws_idx, K);

    const int total = N * D_DIM;
    cvq_recon_kernel<<<total / 256, 256, 0, stream>>>(ws_idx, cb_re, cb_im, out + N);
}